// DecagonModel_10325101379834
// MI455X (gfx1250) — compile-verified
//
#include <hip/hip_runtime.h>
#include <hip/hip_bf16.h>

typedef __attribute__((ext_vector_type(2))) float v2f;
typedef __attribute__((ext_vector_type(8))) float v8f;

// ---------------------------------------------------------------------------
// Dense GEMM: hw[N,64] = h[N,K] @ W[K,64] using V_WMMA_F32_16X16X4_F32.
// 256 threads = 8 waves; block covers 32 rows x 64 cols:
//   wave>>2 selects row half (16 rows), wave&3 selects 16-col tile.
// A frag (16x4 f32): lanes 0-15 hold K={k,k+1} for M=lane, lanes 16-31 hold
// K={k+2,k+3} for M=lane-16  -> one aligned float2 load per lane.
// B frag (4x16 f32): mirrored layout, two scalar loads of W.
// C/D (16x16 f32): VGPR r, lane l -> C[M = r + 8*(l>>4)][N = l&15].
// ---------------------------------------------------------------------------
__global__ __launch_bounds__(256)
void gemm_wmma_f32(const float* __restrict__ h, const float* __restrict__ W,
                   float* __restrict__ hw, int N, int K) {
  const int lane = threadIdx.x & 31;
  const int wave = threadIdx.x >> 5;
  const int rowTile = blockIdx.x * 2 + (wave >> 2);
  const int row0 = rowTile * 16;
  if (row0 >= N) return;
  const int col0 = (wave & 3) * 16;
  const int m = lane & 15;          // M (for A) / N (for B,C)
  const int khalf = (lane >> 4) * 2; // lanes 16-31 handle K+2,K+3

  v8f acc = {};
  const float* __restrict__ arow = h + (size_t)(row0 + m) * K;
  const float* __restrict__ wcol = W + col0 + m;

  for (int k = 0; k < K; k += 4) {
    const int kk = k + khalf;
    v2f a = *(const v2f*)(arow + kk);           // A[M=m][kk], A[M=m][kk+1]
    v2f b;
    b.x = wcol[(size_t)kk * 64];                // B[kk][col0+m]
    b.y = wcol[(size_t)(kk + 1) * 64];          // B[kk+1][col0+m]
    acc = __builtin_amdgcn_wmma_f32_16x16x4_f32(
        /*neg_a=*/false, a, /*neg_b=*/false, b,
        /*c_mod=*/(short)0, acc, /*reuse_a=*/false, /*reuse_b=*/false);
  }

  const int rbase = row0 + (lane >> 4) * 8;
#pragma unroll
  for (int r = 0; r < 8; ++r)
    hw[(size_t)(rbase + r) * 64 + col0 + m] = acc[r];
}

// ---------------------------------------------------------------------------
// Sparse scatter-add: acc[rows[e]][f] += vals[e] * hw[cols[e]][f]
// One float per thread; 64 consecutive threads share one edge -> coalesced
// gathers and consecutive-address hardware f32 atomics.
// ---------------------------------------------------------------------------
__global__ __launch_bounds__(256)
void scatter_add(const int* __restrict__ rows, const int* __restrict__ cols,
                 const float* __restrict__ vals, const float* __restrict__ hw,
                 float* __restrict__ acc, int E) {
  int idx = blockIdx.x * blockDim.x + threadIdx.x;
  if (idx >= E * 64) return;
  int e = idx >> 6;
  int f = idx & 63;
  int c = cols[e];
  int r = rows[e];
  float v = vals[e] * hw[(size_t)c * 64 + f];
  unsafeAtomicAdd(&acc[(size_t)r * 64 + f], v);
}

// In-place ReLU, float4-vectorized.
__global__ __launch_bounds__(256)
void relu4(float4* __restrict__ p, int n4) {
  int i = blockIdx.x * blockDim.x + threadIdx.x;
  if (i >= n4) return;
  float4 v = p[i];
  v.x = fmaxf(v.x, 0.f); v.y = fmaxf(v.y, 0.f);
  v.z = fmaxf(v.z, 0.f); v.w = fmaxf(v.w, 0.f);
  p[i] = v;
}

// Layer-2 ReLU + export att[1]*e1 into out columns [64,128).
__global__ __launch_bounds__(256)
void relu_export(float* __restrict__ acc, float* __restrict__ out,
                 const float* __restrict__ att, int C) {
  int i = blockIdx.x * blockDim.x + threadIdx.x;
  if (i >= 2 * C) return;
  float v = fmaxf(acc[i], 0.f);
  acc[i] = v;
  int type = i / C;
  int rem = i - type * C;
  int node = rem >> 6;
  int f = rem & 63;
  out[(size_t)type * (size_t)C * 3 + (size_t)node * 192 + 64 + f] = att[1] * v;
}

// Final: emb = p5 + h1 (no relu); export att[0]*h1 -> cols [0,64),
// att[2]*emb -> cols [128,192).
__global__ __launch_bounds__(256)
void final_export(const float* __restrict__ h1, const float* __restrict__ p5,
                  float* __restrict__ out, const float* __restrict__ att, int C) {
  int i = blockIdx.x * blockDim.x + threadIdx.x;
  if (i >= 2 * C) return;
  float a = h1[i];
  float emb = p5[i] + a;
  int type = i / C;
  int rem = i - type * C;
  int node = rem >> 6;
  int f = rem & 63;
  size_t base = (size_t)type * (size_t)C * 3 + (size_t)node * 192;
  out[base + f] = att[0] * a;
  out[base + 128 + f] = att[2] * emb;
}

extern "C" void kernel_launch(void* const* d_in, const int* in_sizes, int n_in,
                              void* d_out, int out_size, void* d_ws, size_t ws_size,
                              hipStream_t stream) {
  const float* feat0 = (const float*)d_in[0];
  const float* feat1 = (const float*)d_in[1];
  const int*   rows  = (const int*)d_in[2];
  const int*   cols  = (const int*)d_in[3];
  const float* vals  = (const float*)d_in[4];
  const float* Wl[5] = {(const float*)d_in[5], (const float*)d_in[6],
                        (const float*)d_in[7], (const float*)d_in[8],
                        (const float*)d_in[9]};
  const float* att   = (const float*)d_in[10];

  const int N = in_sizes[0] / 128;   // 100000
  const int E = in_sizes[2] / 4;     // 1000000
  const size_t C = (size_t)N * 64;   // per-type activation size

  float* ws   = (float*)d_ws;
  float* h1   = ws;            // [2C] layer-1 activations (persist for residual)
  float* bufA = ws + 2 * C;    // [2C]
  float* bufB = ws + 4 * C;    // [2C]
  float* hw   = ws + 6 * C;    // [C] per-edge-type GEMM output
  float* out  = (float*)d_out;

  const dim3 blk(256);
  const int gemmBlocks = (N + 31) / 32;
  const int scatBlocks = (int)(((long long)E * 64 + 255) / 256);
  const int elemBlocks = (int)((2 * C + 255) / 256);
  const int relu4Blocks = (int)((2 * C / 4 + 255) / 256);
  const int Ci = (int)C;

  // One hetero GCN layer: acc01 = [acc_type0 | acc_type1], zeroed, then for
  // each edge type t=(i,j): hw = h_j @ W[t]; acc_i += scatter(hw).
  auto layer = [&](const float* in0, const float* in1, const float* W, int K,
                   float* acc01) {
    hipMemsetAsync(acc01, 0, 2 * C * sizeof(float), stream);
    for (int t = 0; t < 4; ++t) {
      const float* hj = (t & 1) ? in1 : in0;
      gemm_wmma_f32<<<gemmBlocks, blk, 0, stream>>>(
          hj, W + (size_t)t * K * 64, hw, N, K);
      float* acc = acc01 + (size_t)(t >> 1) * C;
      scatter_add<<<scatBlocks, blk, 0, stream>>>(
          rows + (size_t)t * E, cols + (size_t)t * E, vals + (size_t)t * E,
          hw, acc, E);
    }
  };

  // Layer 1: feats (K=128) -> h1, relu
  layer(feat0, feat1, Wl[0], 128, h1);
  relu4<<<relu4Blocks, blk, 0, stream>>>((float4*)h1, (int)(2 * C / 4));

  // Layer 2: h1 -> e1 (bufA), relu + export att[1]*e1
  layer(h1, h1 + C, Wl[1], 64, bufA);
  relu_export<<<elemBlocks, blk, 0, stream>>>(bufA, out, att, Ci);

  // Layer 3: e1 -> e2 (bufB), relu
  layer(bufA, bufA + C, Wl[2], 64, bufB);
  relu4<<<relu4Blocks, blk, 0, stream>>>((float4*)bufB, (int)(2 * C / 4));

  // Layer 4: e2 -> e3 (bufA, e1 already exported), relu
  layer(bufB, bufB + C, Wl[3], 64, bufA);
  relu4<<<relu4Blocks, blk, 0, stream>>>((float4*)bufA, (int)(2 * C / 4));

  // Layer 5: e3 -> p5 (bufB), NO relu; emb = p5 + h1
  layer(bufA, bufA + C, Wl[4], 64, bufB);

  // Export att[0]*h1 and att[2]*(p5+h1)
  final_export<<<elemBlocks, blk, 0, stream>>>(h1, bufB, out, att, Ci);

  (void)n_in; (void)out_size; (void)ws_size;
}